// AttributeDecoder_82197084110900
// MI455X (gfx1250) — compile-verified
//
#include <hip/hip_runtime.h>
#include <math.h>

#define HIDF 128
#define NEG_SLOPE 0.2f
#define BN_EPS 1e-5f

typedef __attribute__((ext_vector_type(2))) float v2f;
typedef __attribute__((ext_vector_type(8))) float v8f;

// ---------------------------------------------------------------------------
// GEMM: H[Nx128] = A[Nx128] @ W[128x128] using V_WMMA_F32_16X16X4_F32.
// Block = 256 threads = 8 waves. Block handles a 16-row tile of A; wave w
// handles the 16-column tile [16w, 16w+16). A tile staged in LDS (pad 132).
// ---------------------------------------------------------------------------
__global__ void gemm_wmma_f32(const float* __restrict__ A,
                              const float* __restrict__ W,
                              float* __restrict__ H) {
    __shared__ float As[16 * 132];
    const int row0 = blockIdx.x * 16;
    const int tid  = threadIdx.x;

    // Cooperative load of 16x128 A tile: 2048 floats / 256 threads = 8 each.
    {
        const int r = tid >> 4;            // 0..15
        const int c = (tid & 15) * 8;      // 0..120 step 8
        const float4* g = (const float4*)(A + (size_t)(row0 + r) * HIDF + c);
        float4 v0 = g[0];
        float4 v1 = g[1];
        float* s = &As[r * 132 + c];
        *(float4*)(s)     = v0;
        *(float4*)(s + 4) = v1;
    }
    __syncthreads();

    const int wave = tid >> 5;             // 0..7 -> column tile
    const int lane = tid & 31;
    const int half = lane >> 4;            // 0 or 1
    const int lm   = lane & 15;
    const int ncol = wave * 16 + lm;

    v8f acc = {};
    #pragma unroll
    for (int k = 0; k < HIDF; k += 4) {
        const int ka = k + 2 * half;
        v2f a, b;
        // A 16x4 f32 layout: v0 = K={0|2}, v1 = K={1|3}, M = lane%16
        a.x = As[lm * 132 + ka];
        a.y = As[lm * 132 + ka + 1];
        // B 4x16 f32 layout (mirror): v0 = K={0|2} row, N = lane%16
        b.x = W[(size_t)ka * HIDF + ncol];
        b.y = W[(size_t)(ka + 1) * HIDF + ncol];
        acc = __builtin_amdgcn_wmma_f32_16x16x4_f32(
            false, a, false, b, (short)0, acc, false, false);
    }

    // D layout: VGPR r holds row (r + 8*half), col = lane%16
    #pragma unroll
    for (int r = 0; r < 8; ++r) {
        H[(size_t)(row0 + r + 8 * half) * HIDF + ncol] = acc[r];
    }
}

// ---------------------------------------------------------------------------
// Per-node attention projections: a_s[i] = h[i]·att_src, a_d[i] = h[i]·att_dst
// One wave (32 lanes) per node, 4 floats per lane.
// ---------------------------------------------------------------------------
__global__ void attn_proj(const float* __restrict__ H,
                          const float* __restrict__ att_s,
                          const float* __restrict__ att_d,
                          float* __restrict__ a_s,
                          float* __restrict__ a_d, int n) {
    const int wid  = (blockIdx.x * blockDim.x + threadIdx.x) >> 5;
    const int lane = threadIdx.x & 31;
    if (wid >= n) return;
    const float4 h4 = *(const float4*)(H + (size_t)wid * HIDF + lane * 4);
    const float4 s4 = *(const float4*)(att_s + lane * 4);
    const float4 d4 = *(const float4*)(att_d + lane * 4);
    float ss = h4.x * s4.x + h4.y * s4.y + h4.z * s4.z + h4.w * s4.w;
    float dd = h4.x * d4.x + h4.y * d4.y + h4.z * d4.z + h4.w * d4.w;
    #pragma unroll
    for (int off = 16; off > 0; off >>= 1) {
        ss += __shfl_down(ss, off);
        dd += __shfl_down(dd, off);
    }
    if (lane == 0) { a_s[wid] = ss; a_d[wid] = dd; }
}

__device__ __forceinline__ float leaky(float v) {
    return v > 0.f ? v : NEG_SLOPE * v;
}

// Order-preserving float atomic-max via int/uint atomics (works from -inf).
__device__ __forceinline__ void atomicMaxF(float* addr, float val) {
    if (val >= 0.f) atomicMax((int*)addr, __float_as_int(val));
    else            atomicMin((unsigned int*)addr, __float_as_uint(val));
}

__device__ __forceinline__ void edge_ends(const int* __restrict__ ei, int e,
                                          int E, int& s, int& d) {
    if (e < E) { s = ei[e]; d = ei[E + e]; }
    else       { s = e - E; d = e - E; }          // self loops
}

// Pass 1: segment max of leaky_relu(a_s[src] + a_d[dst]) over dst
__global__ void edge_max(const float* __restrict__ a_s,
                         const float* __restrict__ a_d,
                         const int* __restrict__ ei,
                         float* __restrict__ mx, int E, int Etot) {
    const int e = blockIdx.x * blockDim.x + threadIdx.x;
    if (e >= Etot) return;
    int s, d; edge_ends(ei, e, E, s, d);
    atomicMaxF(&mx[d], leaky(a_s[s] + a_d[d]));
}

// Pass 2: ex = exp(e - max[dst]); denom[dst] += ex
__global__ void edge_exp(const float* __restrict__ a_s,
                         const float* __restrict__ a_d,
                         const int* __restrict__ ei,
                         const float* __restrict__ mx,
                         float* __restrict__ ex,
                         float* __restrict__ dn, int E, int Etot) {
    const int e = blockIdx.x * blockDim.x + threadIdx.x;
    if (e >= Etot) return;
    int s, d; edge_ends(ei, e, E, s, d);
    const float v = __expf(leaky(a_s[s] + a_d[d]) - mx[d]);
    ex[e] = v;
    atomicAdd(&dn[d], v);
}

// Pass 3: out[dst] += (ex/denom[dst]) * h[src]; one wave per edge, float4/lane
__global__ void edge_scatter(const float* __restrict__ H,
                             const float* __restrict__ ex,
                             const float* __restrict__ dn,
                             const int* __restrict__ ei,
                             float* __restrict__ out, int E, int Etot) {
    const int wid  = (blockIdx.x * blockDim.x + threadIdx.x) >> 5;
    const int lane = threadIdx.x & 31;
    if (wid >= Etot) return;
    int s, d; edge_ends(ei, wid, E, s, d);
    const float alpha = ex[wid] / (dn[d] + 1e-16f);
    const float4 h4 = *(const float4*)(H + (size_t)s * HIDF + lane * 4);
    float* o = out + (size_t)d * HIDF + lane * 4;
    atomicAdd(o + 0, alpha * h4.x);
    atomicAdd(o + 1, alpha * h4.y);
    atomicAdd(o + 2, alpha * h4.z);
    atomicAdd(o + 3, alpha * h4.w);
}

// bias + ReLU + BatchNorm(eval) fused
__global__ void post_bn(const float* __restrict__ agg,
                        const float* __restrict__ b,
                        const float* __restrict__ gamma,
                        const float* __restrict__ beta,
                        const float* __restrict__ mean,
                        const float* __restrict__ var,
                        float* __restrict__ out, int total) {
    const int i = blockIdx.x * blockDim.x + threadIdx.x;
    if (i >= total) return;
    const int f = i & (HIDF - 1);
    float v = fmaxf(agg[i] + b[f], 0.f);
    out[i] = (v - mean[f]) * rsqrtf(var[f] + BN_EPS) * gamma[f] + beta[f];
}

// bias + ReLU in place (final output)
__global__ void post_relu(float* __restrict__ io, const float* __restrict__ b,
                          int total) {
    const int i = blockIdx.x * blockDim.x + threadIdx.x;
    if (i >= total) return;
    io[i] = fmaxf(io[i] + b[i & (HIDF - 1)], 0.f);
}

__global__ void fill_f32(float* __restrict__ p, float v, int n) {
    const int i = blockIdx.x * blockDim.x + threadIdx.x;
    if (i < n) p[i] = v;
}

// ---------------------------------------------------------------------------
extern "C" void kernel_launch(void* const* d_in, const int* in_sizes, int n_in,
                              void* d_out, int out_size, void* d_ws, size_t ws_size,
                              hipStream_t stream) {
    const float* x     = (const float*)d_in[0];
    const int*   ei    = (const int*)  d_in[1];
    const float* W1    = (const float*)d_in[2];
    const float* asrc1 = (const float*)d_in[3];
    const float* adst1 = (const float*)d_in[4];
    const float* b1    = (const float*)d_in[5];
    const float* bng   = (const float*)d_in[6];
    const float* bnb   = (const float*)d_in[7];
    const float* bnm   = (const float*)d_in[8];
    const float* bnv   = (const float*)d_in[9];
    const float* W2    = (const float*)d_in[10];
    const float* asrc2 = (const float*)d_in[11];
    const float* adst2 = (const float*)d_in[12];
    const float* b2    = (const float*)d_in[13];
    float* out = (float*)d_out;

    const int N    = in_sizes[0] / HIDF;   // 50000
    const int E    = in_sizes[1] / 2;      // 800000
    const int Etot = E + N;                // + self loops
    const int NF   = N * HIDF;

    // Workspace layout (floats)
    float* ws  = (float*)d_ws;
    float* h1  = ws;                 // N*128   node features (layer in / bn out)
    float* agg = ws + (size_t)NF;    // N*128   GEMM out / scatter accumulator
    float* a_s = ws + 2 * (size_t)NF;
    float* a_d = a_s + N;
    float* mx  = a_d + N;
    float* dn  = mx + N;
    float* ex  = dn + N;             // Etot

    const int T = 256;
    const dim3 blkT(T);
    const dim3 gGemm((N + 15) / 16);
    const dim3 gNode(((size_t)N * 32 + T - 1) / T);     // wave per node
    const dim3 gEdge((Etot + T - 1) / T);               // thread per edge
    const dim3 gScat(((size_t)Etot * 32 + T - 1) / T);  // wave per edge
    const dim3 gElem((NF + T - 1) / T);
    const dim3 gN((N + T - 1) / T);

    // ---------------- Layer 1 ----------------
    fill_f32<<<gElem, blkT, 0, stream>>>(agg, 0.f, NF);
    fill_f32<<<gN,    blkT, 0, stream>>>(mx, -INFINITY, N);
    fill_f32<<<gN,    blkT, 0, stream>>>(dn, 0.f, N);

    gemm_wmma_f32<<<gGemm, blkT, 0, stream>>>(x, W1, h1);
    attn_proj<<<gNode, blkT, 0, stream>>>(h1, asrc1, adst1, a_s, a_d, N);
    edge_max<<<gEdge, blkT, 0, stream>>>(a_s, a_d, ei, mx, E, Etot);
    edge_exp<<<gEdge, blkT, 0, stream>>>(a_s, a_d, ei, mx, ex, dn, E, Etot);
    edge_scatter<<<gScat, blkT, 0, stream>>>(h1, ex, dn, ei, agg, E, Etot);
    post_bn<<<gElem, blkT, 0, stream>>>(agg, b1, bng, bnb, bnm, bnv, h1, NF);

    // ---------------- Layer 2 ----------------
    fill_f32<<<gElem, blkT, 0, stream>>>(out, 0.f, NF);
    fill_f32<<<gN,    blkT, 0, stream>>>(mx, -INFINITY, N);
    fill_f32<<<gN,    blkT, 0, stream>>>(dn, 0.f, N);

    gemm_wmma_f32<<<gGemm, blkT, 0, stream>>>(h1, W2, agg);   // agg = h2
    attn_proj<<<gNode, blkT, 0, stream>>>(agg, asrc2, adst2, a_s, a_d, N);
    edge_max<<<gEdge, blkT, 0, stream>>>(a_s, a_d, ei, mx, E, Etot);
    edge_exp<<<gEdge, blkT, 0, stream>>>(a_s, a_d, ei, mx, ex, dn, E, Etot);
    edge_scatter<<<gScat, blkT, 0, stream>>>(agg, ex, dn, ei, out, E, Etot);
    post_relu<<<gElem, blkT, 0, stream>>>(out, b2, NF);
}